// RegularSimilar_25434796327143
// MI455X (gfx1250) — compile-verified
//
#include <hip/hip_runtime.h>
#include <hip/hip_bf16.h>

typedef __attribute__((ext_vector_type(2))) float v2f;
typedef __attribute__((ext_vector_type(8))) float v8f;

#define BATCH   2048
#define D2      256
#define DIM     128
#define NITEMS  100000
#define NROUNDS (NITEMS / 32)   // 3125 rounds of 32 items
#define NCHUNK  64
#define LDS_PITCH 132           // 128 + 4 pad -> conflict-free b64 reads

__device__ __forceinline__ unsigned int order_f32(float f) {
    unsigned int u = __float_as_uint(f);
    return (u & 0x80000000u) ? ~u : (u | 0x80000000u);
}

// ---------------------------------------------------------------- K0: init
__global__ void k0_init(unsigned long long* __restrict__ rowBest) {
    int i = blockIdx.x * 256 + threadIdx.x;
    if (i < BATCH) rowBest[i] = 0ull;
}

// ------------------------------------------- K1: h = feat @ W + b  (WMMA f32)
// 1024 tiles of 16x16, one wave per tile, K = 256 (64 wmma steps).
__global__ __launch_bounds__(256) void k1_gemm_h(const float* __restrict__ feat,
                                                 const float* __restrict__ W,
                                                 const float* __restrict__ bias,
                                                 float* __restrict__ h) {
    const int tid  = threadIdx.x;
    const int lane = tid & 31;
    const int wave = tid >> 5;
    const int wt   = blockIdx.x * 8 + wave;   // 0..1023
    const int mt   = wt >> 3;                 // 0..127 (batch tiles)
    const int nt   = wt & 7;                  // 0..7   (col tiles)
    const int half = (lane >> 4) & 1;
    const int l15  = lane & 15;
    const int off  = half * 2;

    const int arow = mt * 16 + l15;           // A-matrix row (M = lane&15)
    const int bcol = nt * 16 + l15;           // B-matrix col (N = lane&15)
    const float* arp = feat + arow * D2;

    v8f c = {};
    for (int s = 0; s < 64; ++s) {
        const int k0 = 4 * s + off;
        v2f a = *(const v2f*)(arp + k0);                       // global b64
        v2f b;
        b.x = W[k0 * DIM + bcol];
        b.y = W[(k0 + 1) * DIM + bcol];
        c = __builtin_amdgcn_wmma_f32_16x16x4_f32(false, a, false, b,
                                                  (short)0, c, false, false);
    }
    const float bv = bias[bcol];
    const int rbase = mt * 16 + half * 8;
    for (int r = 0; r < 8; ++r)
        h[(rbase + r) * DIM + bcol] = c[r] + bv;
}

// ------------------------------------- K2: per-column BN -> scale/shift
__global__ __launch_bounds__(256) void k2_bnstats(const float* __restrict__ h,
                                                  const float* __restrict__ gamma,
                                                  const float* __restrict__ beta,
                                                  float* __restrict__ scale,
                                                  float* __restrict__ shift) {
    __shared__ float ssum[256], ssq[256];
    const int col = blockIdx.x;
    const int tid = threadIdx.x;
    float s = 0.f, q = 0.f;
    for (int r = tid; r < BATCH; r += 256) {
        float v = h[r * DIM + col];
        s += v; q += v * v;
    }
    ssum[tid] = s; ssq[tid] = q;
    __syncthreads();
    for (int st = 128; st > 0; st >>= 1) {
        if (tid < st) { ssum[tid] += ssum[tid + st]; ssq[tid] += ssq[tid + st]; }
        __syncthreads();
    }
    if (tid == 0) {
        const float mu  = ssum[0] * (1.f / BATCH);
        const float var = ssq[0] * (1.f / BATCH) - mu * mu;
        const float sc  = gamma[col] * rsqrtf(var + 1e-5f);
        scale[col] = sc;
        shift[col] = beta[col] - mu * sc;
    }
}

// ----------------- K3: scores = uf @ items^T fused with per-row argmax (WMMA)
// Software-pipelined: double-buffered 32-item LDS tiles, next round's global
// loads issued right after the (single) barrier so they overlap 64 WMMAs.
// Each lane's 8 accumulators belong to ONE batch row -> per-lane running
// argmax; one packed atomicMax(u64) per lane at the very end.
__global__ __launch_bounds__(256) void k3_score_argmax(const float* __restrict__ h,
                                                       const float* __restrict__ scale,
                                                       const float* __restrict__ shift,
                                                       const float* __restrict__ items,
                                                       unsigned long long* __restrict__ rowBest) {
    __shared__ float ltile[2][32 * LDS_PITCH];   // 2 x (32 items x 128 K, padded)

    const int tid  = threadIdx.x;
    const int lane = tid & 31;
    const int wave = tid >> 5;
    const int half = (lane >> 4) & 1;
    const int l15  = lane & 15;
    const int off  = half * 2;

    const int rowBase = blockIdx.x * 128 + wave * 16;
    const int myrow   = rowBase + l15;        // this lane's batch row (N index)

    // Preload B-fragments: uf[myrow][k] = leaky(h*scale+shift), striped per
    // the WMMA f32 16x16x4 B layout. Register-resident for the whole kernel.
    v2f bf[32];
    {
        const float* hp = h + myrow * DIM;
        #pragma unroll
        for (int s = 0; s < 32; ++s) {
            const int k0 = 4 * s + off;
            v2f hv = *(const v2f*)(hp + k0);
            v2f sc = *(const v2f*)(scale + k0);
            v2f sh = *(const v2f*)(shift + k0);
            float x0 = hv.x * sc.x + sh.x;  x0 = (x0 >= 0.f) ? x0 : 0.01f * x0;
            float x1 = hv.y * sc.y + sh.y;  x1 = (x1 >= 0.f) ? x1 : 0.01f * x1;
            bf[s].x = x0; bf[s].y = x1;
        }
    }

    float    bestv = -3.402823466e38f;
    unsigned besti = 0u;
    float4   rv[4];                           // staging registers (one round)

    // Round j covers items [j*32, j*32+32): 4096 floats = 4 float4 per thread.
    {
        const float* src = items + (size_t)blockIdx.y * 32 * DIM;
        #pragma unroll
        for (int i = 0; i < 4; ++i)
            rv[i] = *(const float4*)(src + (tid + 256 * i) * 4);
    }

    int p = 0;
    for (int j = blockIdx.y; j < NROUNDS; j += NCHUNK) {
        // Commit staged registers to LDS buffer p.
        #pragma unroll
        for (int i = 0; i < 4; ++i) {
            const int g   = (tid + 256 * i) * 4;
            const int row = g >> 7;
            const int k   = g & 127;
            *(float4*)(&ltile[p][row * LDS_PITCH + k]) = rv[i];
        }
        __syncthreads();   // single barrier per round (safe with ping-pong)

        // Kick off next round's global loads: they overlap the 64 WMMAs below.
        const int jn = j + NCHUNK;
        if (jn < NROUNDS) {
            const float* src = items + (size_t)jn * 32 * DIM;
            #pragma unroll
            for (int i = 0; i < 4; ++i)
                rv[i] = *(const float4*)(src + (tid + 256 * i) * 4);
        }
        if (j + 2 * NCHUNK < NROUNDS)   // warm L2 two rounds ahead (16KB/round)
            __builtin_prefetch(items + (size_t)(j + 2 * NCHUNK) * 32 * DIM + tid * 16, 0, 3);

        const int base = j * 32;
        #pragma unroll
        for (int sub = 0; sub < 2; ++sub) {
            v8f c = {};
            const float* ap = &ltile[p][(sub * 16 + l15) * LDS_PITCH];
            #pragma unroll
            for (int s = 0; s < 32; ++s) {
                v2f a = *(const v2f*)(ap + 4 * s + off);        // ds_load_b64
                c = __builtin_amdgcn_wmma_f32_16x16x4_f32(false, a, false, bf[s],
                                                          (short)0, c, false, false);
            }
            // lane holds scores of batch row `myrow` vs 8 consecutive items
            const int mBase = base + sub * 16 + half * 8;
            #pragma unroll
            for (int r = 0; r < 8; ++r) {
                const float v = c[r];
                if (v > bestv) { bestv = v; besti = (unsigned)(mBase + r); }
            }
        }
        p ^= 1;
    }

    // Pack: high = ordered score, low = ~idx so ties pick the LOWEST index.
    const unsigned long long pack =
        ((unsigned long long)order_f32(bestv) << 32) |
        (unsigned long long)(0xFFFFFFFFu - besti);
    atomicMax(&rowBest[myrow], pack);
}

// --------------------------- K4: decode top1, cosine similarity per row
__global__ __launch_bounds__(256) void k4_sim(const float* __restrict__ items,
                                              const int* __restrict__ uid,
                                              const unsigned long long* __restrict__ rowBest,
                                              float* __restrict__ sims,
                                              float* __restrict__ out_top1) {
    const int tid  = threadIdx.x;
    const int lane = tid & 31;
    const int wave = tid >> 5;
    const int r    = blockIdx.x * 8 + wave;

    const unsigned long long p = rowBest[r];
    const unsigned cidx = 0xFFFFFFFFu - (unsigned)(p & 0xFFFFFFFFull);
    const int      oidx = uid[2 * r + 1];

    const float* a  = items + (size_t)oidx * DIM;
    const float* cc = items + (size_t)cidx * DIM;
    float dac = 0.f, daa = 0.f, dcc = 0.f;
    for (int k = lane; k < DIM; k += 32) {
        const float av = a[k], cv = cc[k];
        dac += av * cv; daa += av * av; dcc += cv * cv;
    }
    for (int m = 16; m >= 1; m >>= 1) {
        dac += __shfl_xor(dac, m, 32);
        daa += __shfl_xor(daa, m, 32);
        dcc += __shfl_xor(dcc, m, 32);
    }
    if (lane == 0) {
        const float na = fmaxf(sqrtf(daa), 1e-6f);
        const float nc = fmaxf(sqrtf(dcc), 1e-6f);
        const float sim = (dac / (na * nc) + 1.f) * 0.5f;
        sims[r]     = sim;
        out_top1[r] = (float)cidx;
    }
}

// --------------------------- K5: final deterministic scalar reductions
__global__ __launch_bounds__(256) void k5_reduce(const float* __restrict__ sims,
                                                 float* __restrict__ out) {
    __shared__ float s1[256], s2[256];
    const int tid = threadIdx.x;
    float a = 0.f, b = 0.f;
    for (int i = tid; i < BATCH; i += 256) {
        const float s = sims[i];
        a += s;
        b += fmaxf(s - 0.5f, 0.f);
    }
    s1[tid] = a; s2[tid] = b;
    __syncthreads();
    for (int st = 128; st > 0; st >>= 1) {
        if (tid < st) { s1[tid] += s1[tid + st]; s2[tid] += s2[tid + st]; }
        __syncthreads();
    }
    if (tid == 0) {
        out[BATCH]     = s2[0] * (1.f / BATCH);  // similarity_loss
        out[BATCH + 1] = s1[0] * (1.f / BATCH);  // mean(sim)
    }
}

extern "C" void kernel_launch(void* const* d_in, const int* in_sizes, int n_in,
                              void* d_out, int out_size, void* d_ws, size_t ws_size,
                              hipStream_t stream) {
    const float* feat  = (const float*)d_in[0];   // (2048,256)
    const float* items = (const float*)d_in[1];   // (100000,128)
    const int*   uid   = (const int*)d_in[2];     // (2048,2)
    const float* W     = (const float*)d_in[3];   // (256,128)
    const float* bias  = (const float*)d_in[4];   // (128,)
    const float* gamma = (const float*)d_in[5];   // (128,)
    const float* beta  = (const float*)d_in[6];   // (128,)
    float* out = (float*)d_out;                   // 2048 top1 + loss + mean_sim

    char* ws = (char*)d_ws;
    float*              h       = (float*)ws;                                 // 1 MiB
    float*              scale   = (float*)(ws + 1048576);                     // 512 B
    float*              shift   = (float*)(ws + 1048576 + 512);               // 512 B
    unsigned long long* rowBest = (unsigned long long*)(ws + 1048576 + 1024); // 16 KiB
    float*              sims    = (float*)(ws + 1048576 + 1024 + 16384);      // 8 KiB

    k0_init<<<8, 256, 0, stream>>>(rowBest);
    k1_gemm_h<<<128, 256, 0, stream>>>(feat, W, bias, h);
    k2_bnstats<<<128, 256, 0, stream>>>(h, gamma, beta, scale, shift);
    k3_score_argmax<<<dim3(16, NCHUNK), 256, 0, stream>>>(h, scale, shift, items, rowBest);
    k4_sim<<<256, 256, 0, stream>>>(items, uid, rowBest, sims, out);
    k5_reduce<<<1, 256, 0, stream>>>(sims, out);
}